// ChainCRF_34565896798975
// MI455X (gfx1250) — compile-verified
//
#include <hip/hip_runtime.h>
#include <hip/hip_bf16.h>

#define BB 512
#define TT 1024
#define NN 50
#define ROOT_IDX 0
#define END_IDX 1

typedef __attribute__((ext_vector_type(2))) float v2f;
typedef __attribute__((ext_vector_type(8))) float v8f;

// state index i (0..63) -> per-lane register slot: slot = 8*(i/16) + (i%8)
// (lane half owning state i is bit3 of (i mod 16))
#define STIDX(i) ((((i) >> 4) << 3) | ((i) & 7))

// WGP-scope prefetch (scope 0 pulls the line into all cache levels including
// the WGP-near cache; __builtin_prefetch can only express SE/DEV/SYS scopes).
__device__ __forceinline__ void prefetch_wgp(const void* p) {
#if defined(__AMDGCN__)
  asm volatile("global_prefetch_b8 %0, off" :: "v"(p));
#endif
}

// v_permlane16_swap_b32(X, Y): X.hi <-> Y.lo.  Returns (ra, rb) where
// ra = {X lanes0-15, Y lanes0-15 seen by lanes16-31} and
// rb = {X lanes16-31 seen by lanes0-15, Y lanes16-31}.
__device__ __forceinline__ void pair_swap(float x, float y, float& ra, float& rb) {
#if __has_builtin(__builtin_amdgcn_permlane16_swap)
  typedef unsigned v2u_t __attribute__((ext_vector_type(2)));
  v2u_t r = __builtin_amdgcn_permlane16_swap(__float_as_uint(x), __float_as_uint(y),
                                             false, false);
  ra = __uint_as_float(r[0]);
  rb = __uint_as_float(r[1]);
#else
  const int h = threadIdx.x & 16;
  ra = h ? __shfl_xor(y, 16, 32) : x;
  rb = h ? y : __shfl_xor(x, 16, 32);
#endif
}

// Exchange value with partner lane (lane ^ 16), VALU path.
__device__ __forceinline__ float half_swap(float x) {
#if __has_builtin(__builtin_amdgcn_permlane16_swap)
  typedef unsigned v2u_t __attribute__((ext_vector_type(2)));
  const unsigned xu = __float_as_uint(x);
  v2u_t r = __builtin_amdgcn_permlane16_swap(xu, xu, false, false);
  const unsigned res = (threadIdx.x & 16) ? r[0] : r[1];
  return __uint_as_float(res);
#else
  return __shfl_xor(x, 16, 32);
#endif
}

// Load one feature row slice for this lane: 4 groups of 8 contiguous states,
// group base g = 16*jt + 8*h. 50 floats per row; (row_offset*50 + g) is even,
// so float2 (b64) loads are 8-byte aligned. Out-of-range states -> 0.
__device__ __forceinline__ void load_feat_row(const float* __restrict__ p, int h,
                                              float* __restrict__ f) {
#pragma unroll
  for (int jt = 0; jt < 4; ++jt) {
    const int g = jt * 16 + 8 * h;
    float v[8];
    if (g + 8 <= NN) {
      const float2 x0 = *(const float2*)(p + g + 0);
      const float2 x1 = *(const float2*)(p + g + 2);
      const float2 x2 = *(const float2*)(p + g + 4);
      const float2 x3 = *(const float2*)(p + g + 6);
      v[0] = x0.x; v[1] = x0.y; v[2] = x1.x; v[3] = x1.y;
      v[4] = x2.x; v[5] = x2.y; v[6] = x3.x; v[7] = x3.y;
    } else if (g < NN) {  // g == 48: states 48,49 valid
      const float2 x0 = *(const float2*)(p + g);
      v[0] = x0.x; v[1] = x0.y;
#pragma unroll
      for (int k = 2; k < 8; ++k) v[k] = 0.f;
    } else {
#pragma unroll
      for (int k = 0; k < 8; ++k) v[k] = 0.f;
    }
#pragma unroll
    for (int k = 0; k < 8; ++k) f[jt * 8 + k] = v[k];
  }
}

// Forward CRF scan, one wave per 16-batch tile.
// alpha kept as (csh, w): alpha[b,j] = csh[b] + log w[b,j] (w unnormalized;
// renormalized every 4 steps, which is scale-only and thus valid even for
// mask-frozen batches). Step: G = (w @ E) * exp(feat); if mask: w = G.
// S^T = E^T @ w^T via V_WMMA_F32_16X16X4_F32; padded chunks 13..15 dropped.
__global__ __launch_bounds__(32)
void crf_forward_wmma(const float* __restrict__ feat,
                      const float* __restrict__ trans,
                      const int* __restrict__ mask,
                      float* __restrict__ logz) {
  const int lane = threadIdx.x;
  const int col  = lane & 15;   // batch within tile / j within A tile
  const int h    = lane >> 4;   // lane half
  const int b0   = blockIdx.x * 16;
  const int b    = b0 + col;

  // ---- static A fragments: A[jt][c] holds E^T rows j=16*jt+col, K=4c+2h,+1
  v2f A[4][13];
#pragma unroll
  for (int jt = 0; jt < 4; ++jt) {
    const int j = jt * 16 + col;
#pragma unroll
    for (int c = 0; c < 13; ++c) {
      const int i0 = 4 * c + 2 * h;
      float a0 = 0.f, a1 = 0.f;
      if (j < NN) {
        if (i0 < NN)     a0 = __expf(trans[i0 * NN + j]);
        if (i0 + 1 < NN) a1 = __expf(trans[(i0 + 1) * NN + j]);
      }
      A[jt][c][0] = a0;
      A[jt][c][1] = a1;
    }
  }

  const float* prow = feat + ((size_t)b * TT) * NN;
  // Prefetch cursor: lane L warms row b0 + L%16 at byte offset 128*(L/16);
  // 2 x 128B lines cover a full 200B feature row -> one uniform prefetch/step.
  const float* pfbase = feat + (((size_t)(b0 + col) * TT) * NN) + 32 * h;

  float w[32];
  float csh = 0.f;

  // ---- t = 0: w = exp(alpha0) = exp(trans[ROOT,:]) * exp(feat[:,0,:])
  {
    float f0[32];
    load_feat_row(prow, h, f0);
#pragma unroll
    for (int idx = 0; idx < 32; ++idx) {
      const int j = ((idx >> 3) << 4) + (idx & 7) + 8 * h;
      const float er = (j < NN) ? __expf(trans[ROOT_IDX * NN + j]) : 0.f;
      w[idx] = er * __expf(f0[idx]);
    }
  }

  // One scan step without normalization.
  auto do_step = [&](int t) {
    const float* p = prow + (size_t)t * NN;
    float f[32];
    load_feat_row(p, h, f);  // issued early; consumed after the WMMA phase

    // warm WGP-near caches ~8 rows ahead
    if (t + 8 < TT) prefetch_wgp(pfbase + (size_t)(t + 8) * NN);

    const int mk = mask[(size_t)b * TT + t];

    // B fragments: one permlane16_swap per (cA,cB) pair per VGPR produces both
    // finished fragments (r0 = cA frag, r1 = cB frag).
    // cA (hp=0) = {0,1,4,5,8,9,12}; cB = cA+2 (hp=1) = {2,3,6,7,10,11}.
    float bfA0[7], bfA1[7], bfB0[6], bfB1[6];
#pragma unroll
    for (int m = 0; m < 6; ++m) {
      const int cAm = 4 * (m >> 1) + (m & 1);
      const int cBm = cAm + 2;
      const int I0  = STIDX(4 * cAm);      // cA own slots (h=0)
      const int SA0 = STIDX(4 * cAm + 2);  // cA crossed slots (held by h=0)
      const int SB0 = STIDX(4 * cBm);      // cB crossed slots (held by h=1)
      const int I1  = STIDX(4 * cBm + 2);  // cB own slots (h=1)
      const float X0 = h ? w[SB0] : w[I0];
      const float Y0 = h ? w[I1]  : w[SA0];
      pair_swap(X0, Y0, bfA0[m], bfB0[m]);
      const float X1 = h ? w[SB0 + 1] : w[I0 + 1];
      const float Y1 = h ? w[I1 + 1]  : w[SA0 + 1];
      pair_swap(X1, Y1, bfA1[m], bfB1[m]);
    }
    {  // chunk 12 (cA only): h=1 side of inputs is dead (always-zero states)
      float dump0, dump1;
      pair_swap(w[STIDX(48)], w[STIDX(50)], bfA0[6], dump0);
      pair_swap(w[STIDX(48) + 1], w[STIDX(50) + 1], bfA1[6], dump1);
    }

    v8f acc[4];
#pragma unroll
    for (int jt = 0; jt < 4; ++jt)
      acc[jt] = (v8f){0.f,0.f,0.f,0.f,0.f,0.f,0.f,0.f};

#pragma unroll
    for (int m = 0; m < 7; ++m) {
      const int cAm = (m < 6) ? (4 * (m >> 1) + (m & 1)) : 12;
      {
        v2f bf; bf[0] = bfA0[m]; bf[1] = bfA1[m];
#pragma unroll
        for (int jt = 0; jt < 4; ++jt)
          acc[jt] = __builtin_amdgcn_wmma_f32_16x16x4_f32(
              false, A[jt][cAm], false, bf, (short)0, acc[jt], false, false);
      }
      if (m < 6) {
        const int cBm = cAm + 2;
        v2f bf; bf[0] = bfB0[m]; bf[1] = bfB1[m];
#pragma unroll
        for (int jt = 0; jt < 4; ++jt)
          acc[jt] = __builtin_amdgcn_wmma_f32_16x16x4_f32(
              false, A[jt][cBm], false, bf, (short)0, acc[jt], false, false);
      }
    }

    if (mk) {  // per-batch => consistent across the lane pair
#pragma unroll
      for (int idx = 0; idx < 32; ++idx)
        w[idx] = acc[idx >> 3][idx & 7] * __expf(f[idx]);
    }
  };

  // Scale-only renormalization (valid for frozen batches too).
  auto renorm = [&]() {
    float m16[16];
#pragma unroll
    for (int i = 0; i < 16; ++i) m16[i] = fmaxf(w[i], w[i + 16]);
#pragma unroll
    for (int i = 0; i < 8; ++i) m16[i] = fmaxf(m16[i], m16[i + 8]);
#pragma unroll
    for (int i = 0; i < 4; ++i) m16[i] = fmaxf(m16[i], m16[i + 4]);
    float mx = fmaxf(fmaxf(m16[0], m16[1]), fmaxf(m16[2], m16[3]));
    mx = fmaxf(mx, half_swap(mx));
    const float r = 1.0f / mx;
#pragma unroll
    for (int idx = 0; idx < 32; ++idx) w[idx] *= r;
    csh += __logf(mx);
  };

  // ---- scan t = 1..T-1, renormalizing every 4 steps
  int t = 1;
  for (; t + 3 < TT; t += 4) {
    do_step(t);
    do_step(t + 1);
    do_step(t + 2);
    do_step(t + 3);
    renorm();
  }
  for (; t < TT; ++t) do_step(t);

  // ---- log_z = csh + log( sum_j w_j * exp(trans[j,END]) )  (any w scale)
  float s = 0.f;
#pragma unroll
  for (int idx = 0; idx < 32; ++idx) {
    const int j = ((idx >> 3) << 4) + (idx & 7) + 8 * h;
    if (j < NN) s += w[idx] * __expf(trans[j * NN + END_IDX]);
  }
  s += half_swap(s);
  const float lz = csh + __logf(s);
  if (lane < 16) logz[b] = lz;
}

// Gold-path score + output: one thread per batch element.
__global__ void crf_score(const float* __restrict__ feat,
                          const float* __restrict__ trans,
                          const int* __restrict__ tags,
                          const int* __restrict__ mask,
                          const float* __restrict__ logz,
                          float* __restrict__ out) {
  const int b = blockIdx.x * blockDim.x + threadIdx.x;
  if (b >= BB) return;
  const int* tg = tags + (size_t)b * TT;
  const int* mk = mask + (size_t)b * TT;
  const float* fb = feat + (size_t)b * TT * NN;

  int prev = tg[0];
  float score = trans[ROOT_IDX * NN + prev];
  score += fb[prev] * (float)mk[0];          // emission t=0 (t <= T-2 branch)
  int len = mk[0];
#pragma unroll 4
  for (int t = 1; t < TT; ++t) {
    const int tag = tg[t];
    const float m = (float)mk[t];
    len += mk[t];
    score += trans[prev * NN + tag] * m;     // transition t-1 -> t
    if (t < TT - 1) score += fb[(size_t)t * NN + tag] * m;  // emissions t<=T-2
    prev = tag;
  }
  const int lt = tg[len - 1];
  score += trans[lt * NN + END_IDX] + fb[(size_t)(TT - 1) * NN + lt] * (float)mk[TT - 1];
  out[b] = logz[b] - score;
}

extern "C" void kernel_launch(void* const* d_in, const int* in_sizes, int n_in,
                              void* d_out, int out_size, void* d_ws, size_t ws_size,
                              hipStream_t stream) {
  const float* feat  = (const float*)d_in[0];
  const float* trans = (const float*)d_in[1];
  const int*   tags  = (const int*)d_in[2];
  const int*   mask  = (const int*)d_in[3];
  float* out  = (float*)d_out;
  float* logz = (float*)d_ws;  // B floats of scratch

  hipLaunchKernelGGL(crf_forward_wmma, dim3(BB / 16), dim3(32), 0, stream,
                     feat, trans, mask, logz);
  hipLaunchKernelGGL(crf_score, dim3(BB / 64), dim3(64), 0, stream,
                     feat, trans, tags, mask, logz, out);
}